// VelVector_22539988370035
// MI455X (gfx1250) — compile-verified
//
#include <hip/hip_runtime.h>

typedef __attribute__((ext_vector_type(2))) float v2f;
typedef __attribute__((ext_vector_type(8))) float v8f;

#define WAVES_PER_BLOCK 8
#define BLOCK_THREADS   256
#define MAX_BLOCKS      4096

// ds_swizzle group-of-32: offset = and[4:0] | or[9:5] | xor[14:10]
template<int OFF>
__device__ __forceinline__ float swz(float v) {
    return __int_as_float(__builtin_amdgcn_ds_swizzle(__float_as_int(v), OFF));
}
#define SWAP16 0x401f   // xor=0x10, and=0x1f : exchange lane <-> lane^16

// Single-instruction ReLU: WMMA output is already canonical (RNE, denorms kept),
// so skip the compiler's extra canonicalizing v_max_num(x,x).
__device__ __forceinline__ float relu1(float x) {
    float r;
    asm("v_max_num_f32 %0, 0, %1" : "=v"(r) : "v"(x));
    return r;
}

__global__ __launch_bounds__(BLOCK_THREADS)
void velvec_kernel(const float* __restrict__ pos, const float* __restrict__ nbr,
                   const float* __restrict__ W1,  const float* __restrict__ b1,
                   const float* __restrict__ W2,  const float* __restrict__ b2,
                   const float* __restrict__ W3,  const float* __restrict__ b3,
                   float* __restrict__ out, int N)
{
    // 2 x (16x16 fp32) transpose scratch per wave
    __shared__ float sm[WAVES_PER_BLOCK * 512];

    const int tid  = threadIdx.x;
    const int lane = tid & 31;
    const int wave = tid >> 5;
    const int hi   = lane >> 4;     // upper half of wave
    const int n    = lane & 15;     // N-column / row-in-tile
    float* tile = sm + wave * 512;

    // ---- per-wave constant setup (amortized over the point loop) ----
    float w1[15], bb1[5], w3[5];
#pragma unroll
    for (int i = 0; i < 15; ++i) w1[i] = W1[i];
#pragma unroll
    for (int i = 0; i < 5; ++i)  bb1[i] = b1[i];
#pragma unroll
    for (int i = 0; i < 5; ++i)  w3[i] = W3[i];
    const float bias3 = b3[0];

    // B operand for layer 2: B[k][n] = W2[n*5+k], zero-padded to 4x16.
    // 32-bit B 4x16 layout: VGPR0 = rows K0(lanes0-15)/K2(lanes16-31), VGPR1 = K1/K3.
    auto w2t = [&](int k, int col) -> float {
        return (k < 5 && col < 5) ? W2[col * 5 + k] : 0.0f;
    };
    v2f B2lo, B2hi;
    B2lo.x = w2t(hi ? 2 : 0, n);
    B2lo.y = w2t(hi ? 3 : 1, n);
    B2hi.x = w2t(hi ? 6 : 4, n);   // only k=4 survives; k=6 pads to 0
    B2hi.y = 0.0f;                 // k=5 / k=7 pad
    const float bias2 = (n < 5) ? b2[n] : 0.0f;   // zero pad -> padded C cols stay 0
    v8f cbias;
#pragma unroll
    for (int j = 0; j < 8; ++j) cbias[j] = bias2;

    const int waveGlobal = blockIdx.x * WAVES_PER_BLOCK + wave;
    const int waveStride = gridDim.x * WAVES_PER_BLOCK;

    for (int p = waveGlobal; p < N; p += waveStride) {
        const float px = pos[3 * p + 0];
        const float py = pos[3 * p + 1];
        const float pz = pos[3 * p + 2];

        const int base = (p * 32 + lane) * 3;        // fits in int (max 48e6)
        const float nx = nbr[base + 0];
        const float ny = nbr[base + 1];
        const float nz = nbr[base + 2];
        // prefetch next point's 384B neighbor slice (global_prefetch_b8)
        __builtin_prefetch(nbr + base + 32 * 3, 0, 1);

        // normalized difference vector (eps = 1e-12, as in F.normalize)
        const float dx = nx - px, dy = ny - py, dz = nz - pz;
        const float nr  = sqrtf(fmaf(dx, dx, fmaf(dy, dy, dz * dz)));
        const float inv = 1.0f / fmaxf(nr, 1e-12f);
        const float ndx = dx * inv, ndy = dy * inv, ndz = dz * inv;

        // ---- layer 1 (16x3x5: below WMMA granularity -> per-lane FMAs) ----
        float h[5];
#pragma unroll
        for (int j = 0; j < 5; ++j) {
            float a = fmaf(w1[3 * j + 2], ndz,
                      fmaf(w1[3 * j + 1], ndy,
                      fmaf(w1[3 * j + 0], ndx, bb1[j])));
            h[j] = fmaxf(a, 0.0f);
        }

        // ---- build A operands (32-bit A 16x4 layout) for two 16-row tiles ----
        const float s0 = swz<SWAP16>(h[0]);
        const float s1 = swz<SWAP16>(h[1]);
        const float s2 = swz<SWAP16>(h[2]);
        const float s3 = swz<SWAP16>(h[3]);
        const float s4 = swz<SWAP16>(h[4]);
        v2f A0lo, A0hi, A1lo, A1hi;
        // tile 0: rows = neighbors 0..15
        A0lo.x = hi ? s2   : h[0];  A0lo.y = hi ? s3   : h[1];   // K0/K2, K1/K3
        A0hi.x = hi ? 0.0f : h[4];  A0hi.y = 0.0f;               // K4/K6, K5/K7
        // tile 1: rows = neighbors 16..31
        A1lo.x = hi ? h[2] : s0;    A1lo.y = hi ? h[3] : s1;
        A1hi.x = hi ? 0.0f : s4;    A1hi.y = 0.0f;

        // ---- layer 2 via chained V_WMMA_F32_16X16X4_F32 (K=5 split 4+1) ----
        v8f c0 = __builtin_amdgcn_wmma_f32_16x16x4_f32(false, A0lo, false, B2lo,
                                                       (short)0, cbias, false, false);
        c0 = __builtin_amdgcn_wmma_f32_16x16x4_f32(false, A0hi, false, B2hi,
                                                   (short)0, c0, false, false);
        v8f c1 = __builtin_amdgcn_wmma_f32_16x16x4_f32(false, A1lo, false, B2lo,
                                                       (short)0, cbias, false, false);
        c1 = __builtin_amdgcn_wmma_f32_16x16x4_f32(false, A1hi, false, B2hi,
                                                   (short)0, c1, false, false);

        // ---- ReLU (1-instr) + C->row transpose through private LDS scratch ----
        // C layout: VGPR j, lanes0-15: (M=j, N=lane); lanes16-31: (M=j+8, N=lane-16)
        const int sbase = hi * 8 * 16 + n;
#pragma unroll
        for (int j = 0; j < 8; ++j) {
            tile[sbase + j * 16]       = relu1(c0[j]);
            tile[256 + sbase + j * 16] = relu1(c1[j]);
        }
        // same-wave DS ops are in-order: safe RAW, no barrier needed
        const float* hrow = tile + hi * 256 + n * 16;  // row `lane` of h2
        float w = bias3;
#pragma unroll
        for (int i = 0; i < 5; ++i) w = fmaf(w3[i], hrow[i], w);

        // ---- vel = sum_k nd*w : 32-lane butterfly via ds_swizzle ----
        float cx = ndx * w, cy = ndy * w, cz = ndz * w;
        cx += swz<0x041f>(cx); cy += swz<0x041f>(cy); cz += swz<0x041f>(cz); // xor 1
        cx += swz<0x081f>(cx); cy += swz<0x081f>(cy); cz += swz<0x081f>(cz); // xor 2
        cx += swz<0x101f>(cx); cy += swz<0x101f>(cy); cz += swz<0x101f>(cz); // xor 4
        cx += swz<0x201f>(cx); cy += swz<0x201f>(cy); cz += swz<0x201f>(cz); // xor 8
        cx += swz<0x401f>(cx); cy += swz<0x401f>(cy); cz += swz<0x401f>(cz); // xor 16

        const float vn = sqrtf(fmaf(cx, cx, fmaf(cy, cy, cz * cz)));
        const float vi = 1.0f / fmaxf(vn, 1e-12f);
        if (lane == 0) {
            out[3 * p + 0] = cx * vi;
            out[3 * p + 1] = cy * vi;
            out[3 * p + 2] = cz * vi;
        }
    }
}

extern "C" void kernel_launch(void* const* d_in, const int* in_sizes, int n_in,
                              void* d_out, int out_size, void* d_ws, size_t ws_size,
                              hipStream_t stream) {
    const float* pos = (const float*)d_in[0];
    const float* nbr = (const float*)d_in[1];
    const float* W1  = (const float*)d_in[2];
    const float* b1  = (const float*)d_in[3];
    const float* W2  = (const float*)d_in[4];
    const float* b2  = (const float*)d_in[5];
    const float* W3  = (const float*)d_in[6];
    const float* b3  = (const float*)d_in[7];
    float* out = (float*)d_out;

    const int N = in_sizes[0] / 3;               // 500000 points
    int blocks = (N + WAVES_PER_BLOCK - 1) / WAVES_PER_BLOCK;
    if (blocks > MAX_BLOCKS) blocks = MAX_BLOCKS;
    velvec_kernel<<<blocks, BLOCK_THREADS, 0, stream>>>(pos, nbr, W1, b1, W2, b2,
                                                        W3, b3, out, N);
}